// VectorQuantizerEMA_5463198401259
// MI455X (gfx1250) — compile-verified
//
#include <hip/hip_runtime.h>
#include <hip/hip_bf16.h>

// ---------------------------------------------------------------------------
// VQ-VAE vector quantizer (eval) for MI455X / gfx1250.
//   distances GEMM on V_WMMA_F32_16X16X4_F32 (f32 precision required: the
//   distance matrix is itself an output), async global->LDS double-buffered
//   tile staging (ASYNCcnt path), non-temporal streaming stores for the
//   ~600MB of outputs.
// ---------------------------------------------------------------------------

typedef __attribute__((ext_vector_type(2))) float v2f;
typedef __attribute__((ext_vector_type(8))) float v8f;

#define N_ROWS 65536      // 32 * 2048
#define DIM    256
#define K_EMB  1024

// flat f32 offsets into d_out (tuple concatenated in return order)
#define OFF_LOSS  0ull
#define OFF_QUANT 1ull
#define OFF_PERP  16777217ull
#define OFF_ENC   16777218ull
#define OFF_DIST  83886082ull
#define OFF_IDX   150994946ull

#define LDS_TILE (64 * 36)   // 64 rows x 32 cols, padded stride 36

// order-preserving f32 -> u32 key, packed over column index so that u64-min
// reproduces argmin with lowest-index tie-break (jnp.argmin semantics).
__device__ __forceinline__ unsigned long long packDist(float d, unsigned int c) {
  unsigned int b = __float_as_uint(d);
  b ^= (b >> 31) ? 0xFFFFFFFFu : 0x80000000u;
  return ((unsigned long long)b << 32) | (unsigned long long)c;
}

// async global -> LDS, 16B per lane, tracked on ASYNCcnt (CDNA5 path)
__device__ __forceinline__ void async_ld_b128(unsigned lds_byte_addr,
                                              unsigned gvoff_bytes,
                                              const void* sbase) {
  asm volatile("global_load_async_to_lds_b128 %0, %1, %2"
               :
               : "v"(lds_byte_addr), "v"(gvoff_bytes), "s"(sbase)
               : "memory");
}
__device__ __forceinline__ void wait_async0() {
  asm volatile("s_wait_asynccnt 0x0" ::: "memory");
}

// ---------------------------------------------------------------------------
// Row squared-norms (wave per row). Optionally initializes argmin slots.
// ---------------------------------------------------------------------------
__global__ __launch_bounds__(256) void vq_row_norms(
    const float* __restrict__ M, float* __restrict__ n2,
    unsigned long long* __restrict__ amin) {
  const int lane = threadIdx.x & 31;
  const int wave = threadIdx.x >> 5;
  const int row  = blockIdx.x * 8 + wave;
  const float* r = M + (size_t)row * DIM;
  float s = 0.f;
  #pragma unroll
  for (int j = lane; j < DIM; j += 32) {
    float v = r[j];
    s += v * v;
  }
  #pragma unroll
  for (int m = 16; m >= 1; m >>= 1) s += __shfl_xor(s, m, 32);
  if (lane == 0) {
    n2[row] = s;
    if (amin) amin[row] = ~0ull;
  }
}

__global__ void vq_init(unsigned int* __restrict__ counts,
                        float* __restrict__ lsum) {
  const int i = blockIdx.x * blockDim.x + threadIdx.x;
  if (i < K_EMB) counts[i] = 0u;
  if (i == 0) *lsum = 0.f;
}

// ---------------------------------------------------------------------------
// Zero the 256MB one-hot region (base is 8B aligned -> v2f NT stores).
// ---------------------------------------------------------------------------
__global__ __launch_bounds__(256) void vq_zero_enc(float* __restrict__ enc) {
  v2f* p = (v2f*)enc;
  const size_t total  = (size_t)N_ROWS * K_EMB / 2;  // 33554432 x 8B
  const size_t stride = (size_t)gridDim.x * blockDim.x;
  const v2f z = {0.f, 0.f};
  for (size_t i = (size_t)blockIdx.x * blockDim.x + threadIdx.x; i < total;
       i += stride)
    __builtin_nontemporal_store(z, &p[i]);
}

// ---------------------------------------------------------------------------
// Tiled distance GEMM: block = 4 waves, 64x64 tile of distances.
//   A = inputs [N,D] row-major, B read as E [K,D] rows (B = E^T block).
//   Double-buffered LDS tiles filled with GLOBAL_LOAD_ASYNC_TO_LDS_B128,
//   padded stride 36 -> conflict-free ds_load_b64 fragment reads.
//   8 x 4 WMMA f32 16x16x4 per 32-deep K chunk, v8f accumulators.
// Epilogue: dist = x2 + e2 - 2*acc, NT-stream out, per-row argmin via
// shuffle tree + one u64 atomic-min per (row, tile).
// ---------------------------------------------------------------------------
__global__ __launch_bounds__(128) void vq_gemm_dist(
    const float* __restrict__ X, const float* __restrict__ E,
    const float* __restrict__ x2, const float* __restrict__ e2,
    float* __restrict__ dist, unsigned long long* __restrict__ amin) {
  __shared__ float As[2][LDS_TILE];
  __shared__ float Bs[2][LDS_TILE];

  const int t    = threadIdx.x;
  const int lane = t & 31;
  const int wave = t >> 5;
  const int half = lane >> 4;   // 0: lanes 0-15, 1: lanes 16-31
  const int ln   = lane & 15;
  const int rowBlock = blockIdx.y * 64;
  const int colBlock = blockIdx.x * 64;

  // LDS byte addresses (low 32 bits of the flat pointer == LDS offset)
  const unsigned ldsA0 = (unsigned)(uintptr_t)(void*)&As[0][0];
  const unsigned ldsB0 = (unsigned)(uintptr_t)(void*)&Bs[0][0];

  const int rr = t >> 3;          // 0..15
  const int cc = (t & 7) << 2;    // 0..28

  // stage one 64x32 chunk of A and B into LDS buffer `buf` (async)
  auto stage = [&](int k0, int buf) {
    #pragma unroll
    for (int p = 0; p < 4; ++p) {
      const int r = p * 16 + rr;  // 0..63
      const unsigned lofs = (unsigned)((buf * LDS_TILE + r * 36 + cc) * 4);
      const unsigned gA = (unsigned)(((rowBlock + r) * DIM + k0 + cc) * 4);
      const unsigned gB = (unsigned)(((colBlock + r) * DIM + k0 + cc) * 4);
      async_ld_b128(ldsA0 + lofs, gA, (const void*)X);
      async_ld_b128(ldsB0 + lofs, gB, (const void*)E);
    }
  };

  const v8f vzero = {0.f, 0.f, 0.f, 0.f, 0.f, 0.f, 0.f, 0.f};
  v8f acc[4];
  #pragma unroll
  for (int s = 0; s < 4; ++s) acc[s] = vzero;

  stage(0, 0);
  for (int i = 0; i < 8; ++i) {   // DIM / 32 chunks
    const int cur = i & 1;
    wait_async0();                // our async fills for buffer `cur` landed
    __syncthreads();              // everyone's fills landed
    if (i < 7) stage((i + 1) * 32, cur ^ 1);  // prefetch next while computing

    const float* Ab = &As[cur][0];
    const float* Bb = &Bs[cur][0];
    #pragma unroll
    for (int kk = 0; kk < 32; kk += 4) {
      // A fragment 16x4: lane ln -> M, halves hold K pairs {0,1}/{2,3}
      const v2f a = *(const v2f*)&Ab[(wave * 16 + ln) * 36 + kk + 2 * half];
      #pragma unroll
      for (int s = 0; s < 4; ++s) {
        // B fragment 4x16: lane ln -> N, halves hold K pairs
        const v2f b = *(const v2f*)&Bb[(s * 16 + ln) * 36 + kk + 2 * half];
        acc[s] = __builtin_amdgcn_wmma_f32_16x16x4_f32(
            false, a, false, b, (short)0, acc[s], false, false);
      }
    }
  }

  // epilogue
  float x2r[8];
  #pragma unroll
  for (int v = 0; v < 8; ++v)
    x2r[v] = x2[rowBlock + wave * 16 + v + 8 * half];
  float e2c[4];
  #pragma unroll
  for (int s = 0; s < 4; ++s) e2c[s] = e2[colBlock + s * 16 + ln];

  #pragma unroll
  for (int v = 0; v < 8; ++v) {
    const int row = rowBlock + wave * 16 + v + 8 * half;
    float bestd = 3.4e38f;
    int bestc = 0x7FFFFFFF;
    #pragma unroll
    for (int s = 0; s < 4; ++s) {
      const int col = colBlock + s * 16 + ln;
      const float d = x2r[v] + e2c[s] - 2.0f * acc[s][v];
      __builtin_nontemporal_store(d, &dist[(size_t)row * K_EMB + col]);
      if (d < bestd) { bestd = d; bestc = col; }
    }
    // min over the 16 lanes that share this row (stay within half-wave)
    #pragma unroll
    for (int m = 1; m < 16; m <<= 1) {
      const float od = __shfl_xor(bestd, m, 32);
      const int   oc = __shfl_xor(bestc, m, 32);
      if (od < bestd || (od == bestd && oc < bestc)) { bestd = od; bestc = oc; }
    }
    if (ln == 0) atomicMin(&amin[row], packDist(bestd, (unsigned)bestc));
  }
}

// ---------------------------------------------------------------------------
// Per-row finalize (wave per row): read winner, gather code row ->
// quantized_st, scatter one-hot, histogram, partial commitment loss.
// ---------------------------------------------------------------------------
__global__ __launch_bounds__(256) void vq_finalize(
    const float* __restrict__ X, const float* __restrict__ E,
    const unsigned long long* __restrict__ amin, float* __restrict__ quant,
    float* __restrict__ enc, float* __restrict__ idx_out,
    unsigned int* __restrict__ counts, float* __restrict__ lsum) {
  const int lane = threadIdx.x & 31;
  const int wave = threadIdx.x >> 5;
  const int row  = blockIdx.x * 8 + wave;

  const unsigned long long p = amin[row];
  const int idx = (int)(p & 0xFFFFFFFFull);

  const float* er = E + (size_t)idx * DIM;
  const float* xr = X + (size_t)row * DIM;
  float* qr = quant + (size_t)row * DIM;

  float partial = 0.f;
  #pragma unroll
  for (int j = lane; j < DIM; j += 32) {
    const float q = er[j];
    const float x = xr[j];
    __builtin_nontemporal_store(q, &qr[j]);  // straight-through == quantized
    const float d = q - x;
    partial += d * d;
  }
  #pragma unroll
  for (int m = 16; m >= 1; m >>= 1) partial += __shfl_xor(partial, m, 32);

  if (lane == 0) {
    atomicAdd(lsum, partial);
    idx_out[row] = (float)idx;
    enc[(size_t)row * K_EMB + idx] = 1.0f;
    atomicAdd(&counts[idx], 1u);
  }
}

// ---------------------------------------------------------------------------
// Scalars: vq_loss and perplexity.
// ---------------------------------------------------------------------------
__global__ __launch_bounds__(1024) void vq_scalars(
    const unsigned int* __restrict__ counts, const float* __restrict__ lsum,
    float* __restrict__ out) {
  __shared__ float sh[1024];
  const int k = threadIdx.x;
  const float p = (float)counts[k] * (1.0f / (float)N_ROWS);
  sh[k] = p * __logf(p + 1e-10f);
  __syncthreads();
  for (int s = 512; s > 0; s >>= 1) {
    if (k < s) sh[k] += sh[k + s];
    __syncthreads();
  }
  if (k == 0) {
    out[OFF_LOSS] = 0.25f * (*lsum) / ((float)N_ROWS * (float)DIM);
    out[OFF_PERP] = __expf(-sh[0]);
  }
}

// ---------------------------------------------------------------------------
extern "C" void kernel_launch(void* const* d_in, const int* in_sizes, int n_in,
                              void* d_out, int out_size, void* d_ws,
                              size_t ws_size, hipStream_t stream) {
  (void)in_sizes; (void)n_in; (void)out_size; (void)ws_size;
  const float* X = (const float*)d_in[0];  // [32,2048,256]
  const float* E = (const float*)d_in[1];  // [1024,256]
  float* out = (float*)d_out;

  // workspace carve-out (< 1 MB)
  float* x2 = (float*)d_ws;                                  // N floats
  float* e2 = x2 + N_ROWS;                                   // K floats
  unsigned long long* amin = (unsigned long long*)(e2 + K_EMB);  // N u64 (8B aligned)
  unsigned int* counts = (unsigned int*)(amin + N_ROWS);     // K u32
  float* lsum = (float*)(counts + K_EMB);                    // 1 f32

  vq_row_norms<<<N_ROWS / 8, 256, 0, stream>>>(X, x2, amin);
  vq_row_norms<<<K_EMB / 8, 256, 0, stream>>>(E, e2, nullptr);
  vq_init<<<4, 256, 0, stream>>>(counts, lsum);
  vq_zero_enc<<<4096, 256, 0, stream>>>(out + OFF_ENC);

  dim3 grid(K_EMB / 64, N_ROWS / 64);  // (16, 1024)
  vq_gemm_dist<<<grid, 128, 0, stream>>>(X, E, x2, e2, out + OFF_DIST, amin);

  vq_finalize<<<N_ROWS / 8, 256, 0, stream>>>(
      X, E, amin, out + OFF_QUANT, out + OFF_ENC, out + OFF_IDX, counts, lsum);

  vq_scalars<<<1, 1024, 0, stream>>>(counts, lsum, out);
}